// GCNRegression_69183333204266
// MI455X (gfx1250) — compile-verified
//
#include <hip/hip_runtime.h>
#include <hip/hip_bf16.h>
#include <stdint.h>

// GCN on MI455X (gfx1250): fp32 WMMA (16x16x4) for dense transforms,
// L2-resident atomic scatter for message passing.

typedef float v2f __attribute__((ext_vector_type(2)));
typedef float v8f __attribute__((ext_vector_type(8)));

#define N_NODES 50000
#define N_EDGES 800000
#define IN_DIM  128
#define HID_DIM 128
#define OUT_DIM 64

// ---------------- degree / normalization ----------------
__global__ void k_deg_init(float* __restrict__ deg, int n) {
    int i = blockIdx.x * blockDim.x + threadIdx.x;
    if (i < n) deg[i] = 1.0f;   // +1 self loop
}

__global__ void k_deg_count(const long long* __restrict__ dst,
                            float* __restrict__ deg, int ne) {
    int e = blockIdx.x * blockDim.x + threadIdx.x;
    if (e < ne) atomicAdd(&deg[(int)dst[e]], 1.0f);
}

__global__ void k_deg_rsqrt(float* __restrict__ deg, int n) {
    int i = blockIdx.x * blockDim.x + threadIdx.x;
    if (i < n) deg[i] = rsqrtf(deg[i]);
}

// ---------------- fp32 WMMA GEMM: C[M,N] = A[M,K] @ B[K,N] ----------------
// grid.x = M/16 row tiles; blockDim.x = 32 * (N/16): wave w owns column tile w.
// A tile (16 x K) staged in LDS, padded row stride 132 -> conflict-free
// column reads (bank = (4*row + k) mod 64).
__global__ void k_wmma_gemm_f32(const float* __restrict__ A,
                                const float* __restrict__ B,
                                float* __restrict__ C,
                                int N, int K) {
    __shared__ float As[16][132];
    const int mbase = blockIdx.x * 16;
    const int tid   = threadIdx.x;

    for (int i = tid; i < 16 * K; i += blockDim.x) {
        int r = i / K, c = i % K;
        As[r][c] = A[(size_t)(mbase + r) * K + c];
    }
    __syncthreads();

    const int wave  = tid >> 5;
    const int lane  = tid & 31;
    const int row   = lane & 15;          // M within tile (A) / N within tile (B,C)
    const int khalf = (lane >> 4) * 2;    // lanes 0-15: K=0,1 ; lanes 16-31: K=2,3
    const int col   = wave * 16 + row;

    v8f c8 = {0.f, 0.f, 0.f, 0.f, 0.f, 0.f, 0.f, 0.f};

    for (int kb = 0; kb < K; kb += 4) {
        v2f a, b;
        a.x = As[row][kb + khalf];
        a.y = As[row][kb + khalf + 1];
        b.x = B[(size_t)(kb + khalf)     * N + col];
        b.y = B[(size_t)(kb + khalf + 1) * N + col];
        c8 = __builtin_amdgcn_wmma_f32_16x16x4_f32(
                 /*neg_a=*/false, a, /*neg_b=*/false, b,
                 /*c_mod=*/(short)0, c8, /*reuse_a=*/false, /*reuse_b=*/false);
    }

    const int orow = (lane >> 4) * 8;     // C layout: VGPR r -> M = r (+8 for hi lanes)
    for (int r = 0; r < 8; ++r) {
        C[(size_t)(mbase + orow + r) * N + col] = c8[r];
    }
}

// ---------------- self-loop + bias init: agg = xw * dinv^2 + b ----------------
template<int F>
__global__ void k_selfloop_init(const float* __restrict__ xw,
                                const float* __restrict__ dinv,
                                const float* __restrict__ bias,
                                float* __restrict__ agg, int n) {
    size_t i = (size_t)blockIdx.x * blockDim.x + threadIdx.x;
    if (i >= (size_t)n * F) return;
    int node = (int)(i / F);
    int f    = (int)(i % F);
    float di = dinv[node];
    agg[i] = xw[i] * (di * di) + bias[f];
}

// ---------------- edge scatter: agg[dst] += xw[src] * dinv[src]*dinv[dst] ----
// F/4 threads per edge, float4 granularity; atomics land in L2 (agg fits).
template<int F>
__global__ void k_edge_scatter(const long long* __restrict__ src,
                               const long long* __restrict__ dst,
                               const float* __restrict__ dinv,
                               const float* __restrict__ xw,
                               float* __restrict__ agg, int ne) {
    const int LPE = F / 4;
    long long t = (long long)blockIdx.x * blockDim.x + threadIdx.x;
    int e = (int)(t / LPE);
    int c = (int)(t % LPE);
    if (e >= ne) return;
    int s = (int)src[e];
    int d = (int)dst[e];
    float nrm = dinv[s] * dinv[d];
    const float4 v = *reinterpret_cast<const float4*>(xw + (size_t)s * F + c * 4);
    float* out = agg + (size_t)d * F + c * 4;
    atomicAdd(out + 0, v.x * nrm);
    atomicAdd(out + 1, v.y * nrm);
    atomicAdd(out + 2, v.z * nrm);
    atomicAdd(out + 3, v.w * nrm);
}

__global__ void k_relu_inplace(float* __restrict__ x, size_t n) {
    size_t i = (size_t)blockIdx.x * blockDim.x + threadIdx.x;
    if (i < n) x[i] = fmaxf(x[i], 0.0f);
}

// ---------------- final FC (64 -> 2) + leaky ReLU ----------------
__global__ void k_fc_leaky(const float* __restrict__ h,     // [n, 64]
                           const float* __restrict__ Wfc,   // [64, 2] row-major
                           const float* __restrict__ bfc,   // [2]
                           float* __restrict__ out, int n) {
    int i = blockIdx.x * blockDim.x + threadIdx.x;
    if (i >= n) return;
    const float* row = h + (size_t)i * OUT_DIM;
    float a0 = bfc[0], a1 = bfc[1];
    #pragma unroll
    for (int k = 0; k < OUT_DIM; ++k) {
        float v = row[k];
        a0 += v * Wfc[k * 2 + 0];
        a1 += v * Wfc[k * 2 + 1];
    }
    a0 = (a0 >= 0.f) ? a0 : 0.01f * a0;
    a1 = (a1 >= 0.f) ? a1 : 0.01f * a1;
    out[(size_t)i * 2 + 0] = a0;
    out[(size_t)i * 2 + 1] = a1;
}

// ---------------- launcher ----------------
extern "C" void kernel_launch(void* const* d_in, const int* in_sizes, int n_in,
                              void* d_out, int out_size, void* d_ws, size_t ws_size,
                              hipStream_t stream) {
    const float*     x    = (const float*)d_in[0];
    const long long* eidx = (const long long*)d_in[1];   // int64 in reference
    const float*     W1   = (const float*)d_in[2];
    const float*     b1   = (const float*)d_in[3];
    const float*     W2   = (const float*)d_in[4];
    const float*     b2   = (const float*)d_in[5];
    const float*     Wfc  = (const float*)d_in[6];
    const float*     bfc  = (const float*)d_in[7];
    float*           out  = (float*)d_out;

    const long long* src = eidx;
    const long long* dst = eidx + N_EDGES;

    // Workspace carve-out (~51.5 MB): dinv | XW (25.6MB) | AGG (25.6MB)
    char* ws = (char*)d_ws;
    size_t off = 0;
    float* dinv = (float*)(ws + off); off += ((size_t)N_NODES * 4 + 255) & ~(size_t)255;
    float* XW   = (float*)(ws + off); off += (size_t)N_NODES * HID_DIM * 4;   // xw1, later xw2
    float* AGG  = (float*)(ws + off);                                         // agg1/h1, later agg2

    const int B = 256;

    // degrees -> dinv
    k_deg_init <<<(N_NODES + B - 1) / B, B, 0, stream>>>(dinv, N_NODES);
    k_deg_count<<<(N_EDGES + B - 1) / B, B, 0, stream>>>(dst, dinv, N_EDGES);
    k_deg_rsqrt<<<(N_NODES + B - 1) / B, B, 0, stream>>>(dinv, N_NODES);

    // ---- layer 1: xw1 = x @ W1  (50000x128 @ 128x128) ----
    k_wmma_gemm_f32<<<N_NODES / 16, 32 * (HID_DIM / 16), 0, stream>>>(
        x, W1, XW, HID_DIM, IN_DIM);
    {
        size_t tot = (size_t)N_NODES * HID_DIM;
        k_selfloop_init<HID_DIM><<<(unsigned)((tot + B - 1) / B), B, 0, stream>>>(
            XW, dinv, b1, AGG, N_NODES);
        long long thr = (long long)N_EDGES * (HID_DIM / 4);
        k_edge_scatter<HID_DIM><<<(unsigned)((thr + B - 1) / B), B, 0, stream>>>(
            src, dst, dinv, XW, AGG, N_EDGES);
        k_relu_inplace<<<(unsigned)((tot + B - 1) / B), B, 0, stream>>>(AGG, tot);
    }

    // ---- layer 2: xw2 = h1 @ W2  (50000x128 @ 128x64), reuse XW buffer ----
    k_wmma_gemm_f32<<<N_NODES / 16, 32 * (OUT_DIM / 16), 0, stream>>>(
        AGG, W2, XW, OUT_DIM, HID_DIM);
    {
        size_t tot = (size_t)N_NODES * OUT_DIM;
        k_selfloop_init<OUT_DIM><<<(unsigned)((tot + B - 1) / B), B, 0, stream>>>(
            XW, dinv, b2, AGG, N_NODES);   // AGG reused: h1 already consumed
        long long thr = (long long)N_EDGES * (OUT_DIM / 4);
        k_edge_scatter<OUT_DIM><<<(unsigned)((thr + B - 1) / B), B, 0, stream>>>(
            src, dst, dinv, XW, AGG, N_EDGES);
    }

    // ---- FC + leaky ReLU ----
    k_fc_leaky<<<(N_NODES + B - 1) / B, B, 0, stream>>>(AGG, Wfc, bfc, out, N_NODES);
}